// CapsuleLayer_59682865545849
// MI455X (gfx1250) — compile-verified
//
#include <hip/hip_runtime.h>

// Problem constants
#define BATCH 256
#define IN_UNITS 16
#define CHN 1152
#define NUMU 10
#define USZ 32
#define NITER 3
#define CCHUNK 48   // channels per sj-block; 1152/48 = 24 chunks

typedef float v2f __attribute__((ext_vector_type(2)));
typedef float v8f __attribute__((ext_vector_type(8)));

static __device__ __forceinline__ v8f wmma_f32_16x16x4(v2f a, v2f b, v8f c) {
    // 8 args: (neg_a, A, neg_b, B, c_mod, C, reuse_a, reuse_b)
    return __builtin_amdgcn_wmma_f32_16x16x4_f32(false, a, false, b, (short)0, c,
                                                 false, false);
}

static __device__ __forceinline__ void atomic_add_f32(float* p, float v) {
    unsafeAtomicAdd(p, v);  // native global_atomic_add_f32
}

static __device__ __forceinline__ v8f v8f_zero() {
    v8f z;
    for (int i = 0; i < 8; ++i) z[i] = 0.0f;
    return z;
}

// ---------------------------------------------------------------------------
// c_ij[c,u] = softmax over u of b_ij[c,u]
// ---------------------------------------------------------------------------
__global__ void softmax_kernel(const float* __restrict__ b_ij,
                               float* __restrict__ c_ij) {
    int c = blockIdx.x * blockDim.x + threadIdx.x;
    if (c >= CHN) return;
    float v[NUMU];
    float mx = -1e30f;
    for (int u = 0; u < NUMU; ++u) {
        v[u] = b_ij[c * NUMU + u];
        mx = fmaxf(mx, v[u]);
    }
    float sum = 0.0f;
    for (int u = 0; u < NUMU; ++u) {
        v[u] = __expf(v[u] - mx);
        sum += v[u];
    }
    float inv = 1.0f / sum;
    for (int u = 0; u < NUMU; ++u) c_ij[c * NUMU + u] = v[u] * inv;
}

// ---------------------------------------------------------------------------
// s_j[b,u,s] += sum_{c in chunk} c_ij[c,u] * (W[c,u] @ x[:,:,c])
// Per block: fixed u (blockIdx.y), 48-channel chunk (blockIdx.x), 8 waves.
// Wave w owns b-tiles {2w, 2w+1}; each wave holds 2 s-tiles x 2 b-tiles of
// f32 WMMA accumulators and chains V_WMMA_F32_16X16X4_F32 across c and K=16.
// ---------------------------------------------------------------------------
__global__ void __launch_bounds__(256)
sj_wmma_kernel(const float* __restrict__ x, const float* __restrict__ W,
               const float* __restrict__ c_ij, float* __restrict__ s_j) {
    const int u    = blockIdx.y;
    const int c0   = blockIdx.x * CCHUNK;
    const int wave = threadIdx.x >> 5;
    const int lane = threadIdx.x & 31;
    const int half = lane >> 4;   // 0: lanes 0-15, 1: lanes 16-31
    const int lid  = lane & 15;

    v8f acc[2][2];
    for (int st = 0; st < 2; ++st)
        for (int bt = 0; bt < 2; ++bt) acc[st][bt] = v8f_zero();

    for (int cc = 0; cc < CCHUNK; ++cc) {
        const int c = c0 + cc;
        const float cw = c_ij[c * NUMU + u];
        const float* Wc = W + (size_t)(c * NUMU + u) * (USZ * IN_UNITS);

        // A fragments: A[s,i] = cw * W[c,u,s,i]
        // layout: lane<16 -> K = ks*4 + {0,1}; lane>=16 -> K = ks*4 + {2,3}; M = lid
        v2f afrag[2][4];
        for (int st = 0; st < 2; ++st) {
            const float* wrow = Wc + (size_t)(st * 16 + lid) * IN_UNITS;
            for (int ks = 0; ks < 4; ++ks) {
                const int kb = ks * 4 + half * 2;
                afrag[st][ks][0] = cw * wrow[kb];
                afrag[st][ks][1] = cw * wrow[kb + 1];
            }
        }
        // B fragments: B[i,b] = x[b,i,c]  (x is [B][I][C])
        v2f bfrag[2][4];
        for (int bt = 0; bt < 2; ++bt) {
            const int b = (wave * 2 + bt) * 16 + lid;
            const float* xb = x + (size_t)b * (IN_UNITS * CHN) + c;
            for (int ks = 0; ks < 4; ++ks) {
                const int kb = ks * 4 + half * 2;
                bfrag[bt][ks][0] = xb[(size_t)kb * CHN];
                bfrag[bt][ks][1] = xb[(size_t)(kb + 1) * CHN];
            }
        }
        for (int st = 0; st < 2; ++st)
            for (int bt = 0; bt < 2; ++bt)
                for (int ks = 0; ks < 4; ++ks)
                    acc[st][bt] = wmma_f32_16x16x4(afrag[st][ks], bfrag[bt][ks],
                                                   acc[st][bt]);
    }

    // Write back: C/D layout -> lane<16: M=r, N=lid; lane>=16: M=8+r, N=lid
    for (int st = 0; st < 2; ++st)
        for (int bt = 0; bt < 2; ++bt) {
            const int b = (wave * 2 + bt) * 16 + lid;
            for (int r = 0; r < 8; ++r) {
                const int s = st * 16 + half * 8 + r;
                atomic_add_f32(&s_j[((size_t)b * NUMU + u) * USZ + s],
                               acc[st][bt][r]);
            }
        }
}

// ---------------------------------------------------------------------------
// squash: reduction over the U axis per (b, s)  (faithful to reference axis=2)
// ---------------------------------------------------------------------------
__global__ void squash_kernel(const float* __restrict__ s_j,
                              float* __restrict__ v_j) {
    int idx = blockIdx.x * blockDim.x + threadIdx.x;
    if (idx >= BATCH * USZ) return;
    const int b = idx / USZ;
    const int s = idx % USZ;
    const float* sp = s_j + (size_t)b * NUMU * USZ + s;
    float mag_sq = 0.0f;
    for (int u = 0; u < NUMU; ++u) {
        float t = sp[(size_t)u * USZ];
        mag_sq += t * t;
    }
    const float mag = __builtin_sqrtf(mag_sq);
    const float scale = mag_sq / ((1.0f + mag_sq) * mag);
    float* vp = v_j + (size_t)b * NUMU * USZ + s;
    for (int u = 0; u < NUMU; ++u) vp[(size_t)u * USZ] = sp[(size_t)u * USZ] * scale;
}

// ---------------------------------------------------------------------------
// agreement: b_ij[c,u] += (1/B) * sum_{b,s} u_hat[b,c,u,s] * v_j[b,u,s]
//          = (1/B) * sum_{s,i} W[c,u,s,i] * (V_u^T @ X)[s, i*C+c]
// One wave per (u, 16-wide column tile of flattened (i,c)). P tile computed
// with WMMA (M=s 2 tiles, N=16 cols, K=256 batch), then W-weighted reduction
// fused in-register, finishing with one f32 atomic per lane (two per column) into b_ij.
// ---------------------------------------------------------------------------
__global__ void __launch_bounds__(32)
agree_wmma_kernel(const float* __restrict__ x, const float* __restrict__ W,
                  const float* __restrict__ v_j, float* __restrict__ b_ij) {
    const int u     = blockIdx.y;
    const int ntile = blockIdx.x;          // 0..1151 ; 72 tiles per i
    const int i     = ntile / 72;
    const int c0    = (ntile % 72) * 16;
    const int n0    = ntile * 16;          // == i*1152 + c0 in flat (i,c)
    const int lane  = threadIdx.x & 31;
    const int half  = lane >> 4;
    const int lid   = lane & 15;

    v8f acc0 = v8f_zero();
    v8f acc1 = v8f_zero();

    for (int kk = 0; kk < BATCH / 4; ++kk) {
        const int b0 = kk * 4 + half * 2;  // this lane-half's K rows
        // A[s,b] = v_j[b,u,s]
        const float* vr = v_j + (size_t)b0 * (NUMU * USZ) + (size_t)u * USZ;
        v2f a0, a1, bf;
        a0[0] = vr[lid];
        a0[1] = vr[NUMU * USZ + lid];
        a1[0] = vr[16 + lid];
        a1[1] = vr[NUMU * USZ + 16 + lid];
        // B[b,n] = x_flat[b, n0+n]  (contiguous in n)
        const float* xr = x + (size_t)b0 * (IN_UNITS * CHN) + n0 + lid;
        bf[0] = xr[0];
        bf[1] = xr[IN_UNITS * CHN];
        acc0 = wmma_f32_16x16x4(a0, bf, acc0);
        acc1 = wmma_f32_16x16x4(a1, bf, acc1);
    }

    // Fused W-weighted reduction over s. Column n = lid is split across
    // lanes lid (s in {0..7} u {16..23}) and lid+16 (s in {8..15} u {24..31}).
    const int c = c0 + lid;
    const float* Wc = W + (size_t)(c * NUMU + u) * (USZ * IN_UNITS) + i;
    float partial = 0.0f;
    for (int r = 0; r < 8; ++r) {
        const int s0 = half * 8 + r;         // s-tile 0
        const int s1 = 16 + half * 8 + r;    // s-tile 1
        partial += Wc[(size_t)s0 * IN_UNITS] * acc0[r];
        partial += Wc[(size_t)s1 * IN_UNITS] * acc1[r];
    }
    atomic_add_f32(&b_ij[c * NUMU + u], partial * (1.0f / BATCH));
}

// ---------------------------------------------------------------------------
extern "C" void kernel_launch(void* const* d_in, const int* in_sizes, int n_in,
                              void* d_out, int out_size, void* d_ws, size_t ws_size,
                              hipStream_t stream) {
    (void)in_sizes; (void)n_in; (void)out_size; (void)ws_size;
    const float* x = (const float*)d_in[0];   // (B, I, C)
    const float* W = (const float*)d_in[1];   // (1, C, U, S, I)

    float* ws   = (float*)d_ws;
    float* b_ij = ws;                               // C*U
    float* c_ij = b_ij + CHN * NUMU;                // C*U
    float* s_j  = c_ij + CHN * NUMU;                // B*U*S
    float* v_j  = s_j + BATCH * NUMU * USZ;         // B*U*S

    hipMemsetAsync(b_ij, 0, CHN * NUMU * sizeof(float), stream);

    for (int t = 0; t < NITER; ++t) {
        softmax_kernel<<<(CHN + 127) / 128, 128, 0, stream>>>(b_ij, c_ij);
        hipMemsetAsync(s_j, 0, BATCH * NUMU * USZ * sizeof(float), stream);
        sj_wmma_kernel<<<dim3(CHN / CCHUNK, NUMU), 256, 0, stream>>>(x, W, c_ij, s_j);
        float* vout = (t == NITER - 1) ? (float*)d_out : v_j;
        squash_kernel<<<(BATCH * USZ + 255) / 256, 256, 0, stream>>>(s_j, vout);
        if (t < NITER - 1) {
            agree_wmma_kernel<<<dim3(CHN, NUMU), 32, 0, stream>>>(x, W, v_j, b_ij);
        }
    }
}